// GCN_12927851561510
// MI455X (gfx1250) — compile-verified
//
#include <hip/hip_runtime.h>

typedef __attribute__((ext_vector_type(2))) float v2f;
typedef __attribute__((ext_vector_type(8))) float v8f;

// ---------------------------------------------------------------- degree init
__global__ void k_init_deg(float* __restrict__ deg, int n) {
  int i = blockIdx.x * blockDim.x + threadIdx.x;
  if (i < n) deg[i] = 1.0f;  // self-loop contributes 1 to every node's degree
}

__global__ void k_deg_accum(const int* __restrict__ dst, float* __restrict__ deg, int E) {
  int e = blockIdx.x * blockDim.x + threadIdx.x;
  if (e < E) unsafeAtomicAdd(&deg[dst[e]], 1.0f);
}

__global__ void k_dinv(float* __restrict__ deg, int n) {
  int i = blockIdx.x * blockDim.x + threadIdx.x;
  if (i < n) deg[i] = rsqrtf(deg[i]);  // deg >= 1 always (self-loops)
}

// ------------------------------------------------- layer 1: t1 = x @ W1 (N,2)x(2,8)
// One wave per 16-row tile. V_WMMA_F32_16X16X4_F32, K=2 zero-padded to 4.
__global__ void __launch_bounds__(32)
k_wmma_l1(const float* __restrict__ x, const float* __restrict__ W1,
          const float* __restrict__ dinv, float* __restrict__ t1,
          float* __restrict__ agg1, int n) {
  const int tile = blockIdx.x;
  const int lane = threadIdx.x;
  const int half = lane >> 4;     // 0: K=0,1   1: K=2,3
  const int l    = lane & 15;
  int row = tile * 16 + l;
  int rc  = row < n ? row : n - 1;            // clamp loads, keep EXEC full for WMMA

  // A (16x4 f32): VGPR0 = K=(2*half), VGPR1 = K=(2*half+1); x has only K=0,1
  v2f xv = *(const v2f*)(x + (size_t)rc * 2);  // b64 load
  v2f a;
  a.x = half ? 0.0f : xv.x;
  a.y = half ? 0.0f : xv.y;

  // B (4x16 f32): VGPR0 lanes0-15=K0 row, lanes16-31=K2 row; VGPR1 = K1/K3 rows.
  // W1 is 2x8 row-major -> rows K>=2 and cols >=8 are zero pad.
  // Branch-free: clamped-address loads + selects.
  float w1a = W1[(l & 7)];        // row 0
  float w1b = W1[8 + (l & 7)];    // row 1
  bool  bok = (!half) && (l < 8);
  v2f b;
  b.x = bok ? w1a : 0.0f;
  b.y = bok ? w1b : 0.0f;

  v8f c = {};
  c = __builtin_amdgcn_wmma_f32_16x16x4_f32(false, a, false, b, (short)0, c, false, false);

  // D: vgpr r -> row M=r (lanes 0-15) / M=r+8 (lanes 16-31), col = l
  const bool fullt = (tile * 16 + 16) <= n;   // wave-uniform -> scalar branch
  if (l < 8) {
    const int base = tile * 16 + half * 8;
    if (fullt) {
#pragma unroll
      for (int r = 0; r < 8; ++r) {
        int node = base + r;
        float v  = c[r];
        float di = dinv[node];
        t1[(size_t)node * 8 + l]   = v;
        agg1[(size_t)node * 8 + l] = v * di * di;  // self-loop message, norm = dinv^2
      }
    } else {
#pragma unroll
      for (int r = 0; r < 8; ++r) {
        int node = base + r;
        if (node < n) {
          float v  = c[r];
          float di = dinv[node];
          t1[(size_t)node * 8 + l]   = v;
          agg1[(size_t)node * 8 + l] = v * di * di;
        }
      }
    }
  }
}

// ---------------- layer-1 scatter: one thread per edge, b128 gathers, 8 atomics
__global__ void k_scatter1(const int* __restrict__ src, const int* __restrict__ dst,
                           const float* __restrict__ dinv, const float* __restrict__ t1,
                           float* __restrict__ agg1, int E) {
  int e = blockIdx.x * blockDim.x + threadIdx.x;
  if (e >= E) return;
  int s = src[e], d = dst[e];
  float w = dinv[s] * dinv[d];
  const float4* tp = (const float4*)(t1 + (size_t)s * 8);
  float4 v0 = tp[0];
  float4 v1 = tp[1];
  float* ap = agg1 + (size_t)d * 8;
  unsafeAtomicAdd(ap + 0, v0.x * w);
  unsafeAtomicAdd(ap + 1, v0.y * w);
  unsafeAtomicAdd(ap + 2, v0.z * w);
  unsafeAtomicAdd(ap + 3, v0.w * w);
  unsafeAtomicAdd(ap + 4, v1.x * w);
  unsafeAtomicAdd(ap + 5, v1.y * w);
  unsafeAtomicAdd(ap + 6, v1.z * w);
  unsafeAtomicAdd(ap + 7, v1.w * w);
}

// ---------------- layer 2: t2 = relu(agg1+b1) @ W2 (N,8)x(8,4); two chained WMMAs
__global__ void __launch_bounds__(32)
k_wmma_l2(const float* __restrict__ agg1, const float* __restrict__ b1,
          const float* __restrict__ W2, const float* __restrict__ b2,
          const float* __restrict__ dinv, float* __restrict__ t2,
          float* __restrict__ out, int n) {
  const int tile = blockIdx.x;
  const int lane = threadIdx.x;
  const int half = lane >> 4;
  const int l    = lane & 15;
  int row = tile * 16 + l;
  int rc  = row < n ? row : n - 1;
  const int ka = half * 2;  // this lane's K pair base within each 4-wide K chunk

  // h[row][k] = relu(agg1[row][k] + b1[k]) computed on the fly; paired b64 loads.
  v2f g0 = *(const v2f*)(agg1 + (size_t)rc * 8 + ka);       // k = ka, ka+1
  v2f g1 = *(const v2f*)(agg1 + (size_t)rc * 8 + ka + 4);   // k = ka+4, ka+5
  v2f bb0 = *(const v2f*)(b1 + ka);
  v2f bb1 = *(const v2f*)(b1 + ka + 4);
  v2f a1, a2;
  a1.x = fmaxf(g0.x + bb0.x, 0.0f);
  a1.y = fmaxf(g0.y + bb0.y, 0.0f);
  a2.x = fmaxf(g1.x + bb1.x, 0.0f);
  a2.y = fmaxf(g1.y + bb1.y, 0.0f);

  // W2 is 8x4 row-major; B cols >= 4 are zero pad. Branch-free clamped loads.
  const int lc = l & 3;
  float wr0 = W2[(ka)     * 4 + lc];   // rows 0/2
  float wr1 = W2[(ka + 1) * 4 + lc];   // rows 1/3
  float wr2 = W2[(ka + 4) * 4 + lc];   // rows 4/6
  float wr3 = W2[(ka + 5) * 4 + lc];   // rows 5/7
  bool cok = (l < 4);
  v2f b1v, b2v;
  b1v.x = cok ? wr0 : 0.0f;
  b1v.y = cok ? wr1 : 0.0f;
  b2v.x = cok ? wr2 : 0.0f;
  b2v.y = cok ? wr3 : 0.0f;

  v8f c = {};
  c = __builtin_amdgcn_wmma_f32_16x16x4_f32(false, a1, false, b1v, (short)0, c, false, false);
  c = __builtin_amdgcn_wmma_f32_16x16x4_f32(false, a2, false, b2v, (short)0, c, false, false);

  const bool fullt = (tile * 16 + 16) <= n;
  if (l < 4) {
    const int base = tile * 16 + half * 8;
    float bias = b2[l];
    if (fullt) {
#pragma unroll
      for (int r = 0; r < 8; ++r) {
        int node = base + r;
        float v  = c[r];
        float di = dinv[node];
        t2[(size_t)node * 4 + l]  = v;
        out[(size_t)node * 4 + l] = bias + v * di * di;  // bias + self-loop message
      }
    } else {
#pragma unroll
      for (int r = 0; r < 8; ++r) {
        int node = base + r;
        if (node < n) {
          float v  = c[r];
          float di = dinv[node];
          t2[(size_t)node * 4 + l]  = v;
          out[(size_t)node * 4 + l] = bias + v * di * di;
        }
      }
    }
  }
}

// ---------------- layer-2 scatter: one thread per edge, b128 gather, 4 atomics
__global__ void k_scatter2(const int* __restrict__ src, const int* __restrict__ dst,
                           const float* __restrict__ dinv, const float* __restrict__ t2,
                           float* __restrict__ out, int E) {
  int e = blockIdx.x * blockDim.x + threadIdx.x;
  if (e >= E) return;
  int s = src[e], d = dst[e];
  float w = dinv[s] * dinv[d];
  float4 v = *(const float4*)(t2 + (size_t)s * 4);
  float* op = out + (size_t)d * 4;
  unsafeAtomicAdd(op + 0, v.x * w);
  unsafeAtomicAdd(op + 1, v.y * w);
  unsafeAtomicAdd(op + 2, v.z * w);
  unsafeAtomicAdd(op + 3, v.w * w);
}

extern "C" void kernel_launch(void* const* d_in, const int* in_sizes, int n_in,
                              void* d_out, int out_size, void* d_ws, size_t ws_size,
                              hipStream_t stream) {
  const float* x  = (const float*)d_in[0];
  const int*   ei = (const int*)d_in[1];
  const float* W1 = (const float*)d_in[2];
  const float* b1 = (const float*)d_in[3];
  const float* W2 = (const float*)d_in[4];
  const float* b2 = (const float*)d_in[5];
  float* out = (float*)d_out;

  const int n = in_sizes[0] / 2;   // x is [N,2]
  const int E = in_sizes[1] / 2;   // edge_index is [2,E]
  const int* src = ei;
  const int* dst = ei + E;

  // workspace carve-up (~16.8 MB total, all 16B aligned): dinv | t1 | agg1 | t2
  char* ws = (char*)d_ws;
  size_t nAl = ((size_t)n + 3) & ~(size_t)3;           // keep 16B alignment
  float* dinv = (float*)ws;  ws += sizeof(float) * nAl;
  float* t1   = (float*)ws;  ws += sizeof(float) * (size_t)n * 8;
  float* agg1 = (float*)ws;  ws += sizeof(float) * (size_t)n * 8;
  float* t2   = (float*)ws;

  const int tiles = (n + 15) / 16;

  k_init_deg <<<(n + 255) / 256, 256, 0, stream>>>(dinv, n);
  k_deg_accum<<<(E + 255) / 256, 256, 0, stream>>>(dst, dinv, E);
  k_dinv     <<<(n + 255) / 256, 256, 0, stream>>>(dinv, n);

  k_wmma_l1<<<tiles, 32, 0, stream>>>(x, W1, dinv, t1, agg1, n);

  k_scatter1<<<(E + 255) / 256, 256, 0, stream>>>(src, dst, dinv, t1, agg1, E);

  k_wmma_l2<<<tiles, 32, 0, stream>>>(agg1, b1, W2, b2, dinv, t2, out, n);

  k_scatter2<<<(E + 255) / 256, 256, 0, stream>>>(src, dst, dinv, t2, out, E);
}